// GCN_Reference_35485019799610
// MI455X (gfx1250) — compile-verified
//
#include <hip/hip_runtime.h>
#include <hip/hip_bf16.h>
#include <stddef.h>

// ---------------------------------------------------------------------------
// GCN block for MI455X (gfx1250, wave32, WMMA).
//   y   = sum_s W_s @ (x @ (A_s + alpha*I))  + sum_s b_s      (pre-BN)
//   r   = down_W @ x + down_b                                  (pre-BN)
//   out = relu(BN(y) + BN(r))   (training-mode BN over N,T,V per channel)
//
// fp32 end to end via V_WMMA_F32_16X16X4_F32 (the CDNA5 fp32 matrix path).
// Register-blocked: 1 A-frag feeds 7 (stage2) / 5 (stage1) accumulators,
// x tile staged via GLOBAL_LOAD_ASYNC_TO_LDS_B32 (ASYNCcnt path).
// Pipeline: k_zero -> k_main (GEMMs + stats atomics) -> k_stats -> k_final.
// ---------------------------------------------------------------------------

typedef float v2f __attribute__((ext_vector_type(2)));
typedef float v8f __attribute__((ext_vector_type(8)));

union Acc { v8f v; float f[8]; };

#define AS1 __attribute__((address_space(1)))
#define AS3 __attribute__((address_space(3)))

#if __has_builtin(__builtin_amdgcn_global_load_async_to_lds_b32) && \
    __has_builtin(__builtin_amdgcn_s_wait_asynccnt)
#define HAVE_ASYNC_LDS 1
#else
#define HAVE_ASYNC_LDS 0
#endif

#define TT     4            // time steps per tile
#define KP     28           // V=25 padded to mult of 4 (WMMA K step)
#define COLS   (TT*KP)      // 112 = N-dim column space of stage 2
#define NJ     80           // S*V=75 padded to mult of 16 (stage-1 N dim)
#define ASTR   260          // A_big row stride (256 padded: kills 16-way LDS bank conflict)

// LDS partition (floats)
#define OFF_ABIG   0
#define SZ_ABIG    (128*ASTR)                 // [o=128][k=256] : Wcat | down_W
#define OFF_BBIG   (OFF_ABIG + SZ_ABIG)
#define SZ_BBIG    (256*COLS)                 // rows 0..191 = xa[(s,c)], rows 192..255 = x[c]
#define OFF_ACAT   (OFF_BBIG + SZ_BBIG)
#define SZ_ACAT    (KP*NJ)                    // padded [A_0|A_1|A_2] (+alpha*I)
#define OFF_BSUM   (OFF_ACAT + SZ_ACAT)
#define OFF_DBIA   (OFF_BSUM + 128)
#define OFF_STAT   (OFF_DBIA + 128)
#define LDS_FLOATS (OFF_STAT + 512)
#define LDS_BYTES  (LDS_FLOATS * 4)

__global__ void k_zero(float* __restrict__ g) {
    g[threadIdx.x] = 0.f;
    g[256 + threadIdx.x] = 0.f;
}

__global__ __launch_bounds__(256, 1)
void k_main(const float* __restrict__ x, const float* __restrict__ A,
            const float* __restrict__ W, const float* __restrict__ b,
            const float* __restrict__ alpha, const float* __restrict__ dW,
            const float* __restrict__ db,
            float* __restrict__ y_out, float* __restrict__ r_ws,
            float* __restrict__ gstats)
{
    extern __shared__ float smem[];
    float* A_big = smem + OFF_ABIG;
    float* B_big = smem + OFF_BBIG;
    float* Acat  = smem + OFF_ACAT;
    float* bsum  = smem + OFF_BSUM;
    float* dbias = smem + OFF_DBIA;
    float* stats = smem + OFF_STAT;
    float* xs    = B_big + 192*COLS;     // x tile as [(c,t) x 28] == [c x 112]

    const int tid  = threadIdx.x;
    const int lane = tid & 31, wave = tid >> 5;
    const int lc   = lane & 15, hl = lane >> 4;      // WMMA half-wave select
    const int n    = blockIdx.x / 75;
    const int t0   = (blockIdx.x % 75) * TT;

    // ---- stage 0: fill LDS ------------------------------------------------
    // x tile: rows rho=(c*4+t), 25 valid cols + 3 zero pad.
    // Async DMA straight into LDS (ASYNCcnt path) when available.
    const float* xb = x + (size_t)n * 64 * 300 * 25;
#if HAVE_ASYNC_LDS
    for (int e = tid; e < 256*25; e += 256) {
        int rho = e / 25, v = e - rho*25;
        int c = rho >> 2, t = rho & 3;
        const float* gp = &xb[((size_t)c*300 + (t0 + t))*25 + v];
        float* lp = &xs[rho*KP + v];
        __builtin_amdgcn_global_load_async_to_lds_b32(
            (AS1 int*)gp, (AS3 int*)lp, 0, 0);
    }
#else
    for (int e = tid; e < 256*25; e += 256) {
        int rho = e / 25, v = e - rho*25;
        int c = rho >> 2, t = rho & 3;
        xs[rho*KP + v] = xb[((size_t)c*300 + (t0 + t))*25 + v];
    }
#endif
    for (int p = tid; p < 256*3; p += 256) xs[(p/3)*KP + 25 + (p%3)] = 0.f;
    // A_big[o][k] = W[s][o][c] (k=s*64+c, k<192) | down_W[o][c] (k>=192)
    for (int idx = tid; idx < 128*256; idx += 256) {
        int o = idx >> 8, k = idx & 255;
        float v;
        if (k < 192) { int s = k >> 6, c = k & 63; v = W[((size_t)(s*128 + o))*64 + c]; }
        else          v = dW[o*64 + (k - 192)];
        A_big[o*ASTR + k] = v;
    }
    // Acat[v][s*25+w] = A[s][v][w] + alpha*(v==w), zero padded
    const float al = alpha[0];
    for (int idx = tid; idx < KP*NJ; idx += 256) {
        int v = idx / NJ, j = idx - v*NJ;
        float val = 0.f;
        if (v < 25 && j < 75) {
            int s = j / 25, w = j - s*25;
            val = A[(s*25 + v)*25 + w] + ((v == w) ? al : 0.f);
        }
        Acat[idx] = val;
    }
    // zero xa pad columns (w=25..27) so stage-2 pad outputs stay finite
    for (int p = tid; p < 192*TT*3; p += 256) {
        int row = p / (TT*3), q = p - row*(TT*3);
        B_big[row*COLS + (q/3)*KP + 25 + (q%3)] = 0.f;
    }
    if (tid < 128) {
        bsum[tid]  = b[tid] + b[128 + tid] + b[256 + tid];
        dbias[tid] = db[tid];
    }
    for (int k = tid; k < 512; k += 256) stats[k] = 0.f;
#if HAVE_ASYNC_LDS
    __builtin_amdgcn_s_wait_asynccnt(0);
#endif
    __syncthreads();

    // ---- stage 1: xa[(s,c)][(t,w)] = x @ Acat   (M=256, K=28, N=80) -------
    // 5-wide N register blocking: one A-frag feeds 5 accumulators.
    for (int mt2 = 0; mt2 < 2; ++mt2) {
        const int mt = wave*2 + mt2;
        const int m  = mt*16 + lc;
        Acc acc[5] = {};
        #pragma unroll
        for (int kk = 0; kk < KP; kk += 4) {
            int k0 = kk + 2*hl;
            v2f a;
            a.x = xs[m*KP + k0];  a.y = xs[m*KP + k0 + 1];
            #pragma unroll
            for (int nt = 0; nt < 5; ++nt) {
                int nn = nt*16 + lc;
                v2f bb;
                bb.x = Acat[k0*NJ + nn];  bb.y = Acat[(k0 + 1)*NJ + nn];
                acc[nt].v = __builtin_amdgcn_wmma_f32_16x16x4_f32(
                                false, a, false, bb, (short)0, acc[nt].v, false, false);
            }
        }
        #pragma unroll
        for (int nt = 0; nt < 5; ++nt) {
            int j = nt*16 + lc;                  // column in (s,w) space
            if (j < 75) {
                int s = j / 25, w = j - s*25;
                #pragma unroll
                for (int i = 0; i < 8; ++i) {
                    int rho = mt*16 + i + 8*hl;  // (c,t) row
                    int c = rho >> 2, t = rho & 3;
                    B_big[(s*64 + c)*COLS + t*KP + w] = acc[nt].f[i];
                }
            }
        }
    }
    __syncthreads();

    // ---- stage 2: y = Wcat @ xa (K=0..191) ; r = down_W @ x (K=192..255) --
    // Each wave owns one 16-row o-tile; 7-wide N register blocking.
    const int ot = wave;
    const int oa = ot*16 + lc;
    const size_t outbase = (size_t)n * 128 * 7500;

    {   // y pass
        Acc ay[7] = {};
        for (int kk = 0; kk < 192; kk += 4) {
            int k0 = kk + 2*hl;
            v2f a;
            a.x = A_big[oa*ASTR + k0];  a.y = A_big[oa*ASTR + k0 + 1];
            #pragma unroll
            for (int ct = 0; ct < 7; ++ct) {
                int nn = ct*16 + lc;
                v2f bb;
                bb.x = B_big[k0*COLS + nn];  bb.y = B_big[(k0 + 1)*COLS + nn];
                ay[ct].v = __builtin_amdgcn_wmma_f32_16x16x4_f32(
                               false, a, false, bb, (short)0, ay[ct].v, false, false);
            }
        }
        #pragma unroll
        for (int ct = 0; ct < 7; ++ct) {
            const int col = ct*16 + lc;
            const int t = col / KP, w = col - t*KP;
            const bool valid = (w < 25);
            #pragma unroll
            for (int i = 0; i < 8; ++i) {
                int o = ot*16 + i + 8*hl;
                float yv = ay[ct].f[i] + bsum[o];
                if (valid) {
                    size_t gi = outbase + (size_t)o*7500 + (size_t)(t0 + t)*25 + w;
                    y_out[gi] = yv;
                }
                float sy  = valid ? yv      : 0.f;
                float sy2 = valid ? yv * yv : 0.f;
                #pragma unroll
                for (int md = 1; md < 16; md <<= 1) {
                    sy  += __shfl_xor(sy,  md, 16);
                    sy2 += __shfl_xor(sy2, md, 16);
                }
                if (lc == 0) {
                    atomicAdd(&stats[      o], sy);
                    atomicAdd(&stats[128 + o], sy2);
                }
            }
        }
    }
    {   // r pass (residual down-projection, K rows 192..255 are the raw x tile)
        Acc ar[7] = {};
        for (int kk = 192; kk < 256; kk += 4) {
            int k0 = kk + 2*hl;
            v2f a;
            a.x = A_big[oa*ASTR + k0];  a.y = A_big[oa*ASTR + k0 + 1];
            #pragma unroll
            for (int ct = 0; ct < 7; ++ct) {
                int nn = ct*16 + lc;
                v2f bb;
                bb.x = B_big[k0*COLS + nn];  bb.y = B_big[(k0 + 1)*COLS + nn];
                ar[ct].v = __builtin_amdgcn_wmma_f32_16x16x4_f32(
                               false, a, false, bb, (short)0, ar[ct].v, false, false);
            }
        }
        #pragma unroll
        for (int ct = 0; ct < 7; ++ct) {
            const int col = ct*16 + lc;
            const int t = col / KP, w = col - t*KP;
            const bool valid = (w < 25);
            #pragma unroll
            for (int i = 0; i < 8; ++i) {
                int o = ot*16 + i + 8*hl;
                float rv = ar[ct].f[i] + dbias[o];
                if (valid) {
                    size_t gi = outbase + (size_t)o*7500 + (size_t)(t0 + t)*25 + w;
                    r_ws[gi] = rv;
                }
                float sr  = valid ? rv      : 0.f;
                float sr2 = valid ? rv * rv : 0.f;
                #pragma unroll
                for (int md = 1; md < 16; md <<= 1) {
                    sr  += __shfl_xor(sr,  md, 16);
                    sr2 += __shfl_xor(sr2, md, 16);
                }
                if (lc == 0) {
                    atomicAdd(&stats[256 + o], sr);
                    atomicAdd(&stats[384 + o], sr2);
                }
            }
        }
    }
    __syncthreads();
    for (int k = tid; k < 512; k += 256) atomicAdd(&gstats[k], stats[k]);
}

__global__ void k_stats(const float* __restrict__ gstats,
                        const float* __restrict__ gamma, const float* __restrict__ beta,
                        const float* __restrict__ dgamma, const float* __restrict__ dbeta,
                        float* __restrict__ coef)
{
    int o = threadIdx.x;
    if (o < 128) {
        const float inv_cnt = 1.f / 480000.f;      // N*T*V
        float m1 = gstats[o] * inv_cnt;
        float v1 = gstats[128 + o] * inv_cnt - m1*m1;
        float a1 = gamma[o] * rsqrtf(v1 + 1e-5f);
        float c1 = beta[o] - a1*m1;
        float m2 = gstats[256 + o] * inv_cnt;
        float v2 = gstats[384 + o] * inv_cnt - m2*m2;
        float a2 = dgamma[o] * rsqrtf(v2 + 1e-5f);
        float c2 = dbeta[o] - a2*m2;
        coef[o] = a1; coef[128 + o] = c1; coef[256 + o] = a2; coef[384 + o] = c2;
    }
}

__global__ __launch_bounds__(256)
void k_final(float* __restrict__ y, const float* __restrict__ r,
             const float* __restrict__ coef)
{
    size_t i = (size_t)blockIdx.x * 256 + threadIdx.x;     // float4 index
    size_t f = i * 4;
    int o = (int)((f / 7500) % 128);                       // 7500 % 4 == 0
    float a1 = coef[o], c1 = coef[128 + o];
    float a2 = coef[256 + o], c2 = coef[384 + o];
    float4 yv = ((const float4*)y)[i];
    float4 rv = ((const float4*)r)[i];
    float4 out;
    out.x = fmaxf(a1*yv.x + c1 + a2*rv.x + c2, 0.f);
    out.y = fmaxf(a1*yv.y + c1 + a2*rv.y + c2, 0.f);
    out.z = fmaxf(a1*yv.z + c1 + a2*rv.z + c2, 0.f);
    out.w = fmaxf(a1*yv.w + c1 + a2*rv.w + c2, 0.f);
    ((float4*)y)[i] = out;
}

extern "C" void kernel_launch(void* const* d_in, const int* in_sizes, int n_in,
                              void* d_out, int out_size, void* d_ws, size_t ws_size,
                              hipStream_t stream)
{
    const float* x      = (const float*)d_in[0];
    const float* A      = (const float*)d_in[1];
    const float* W      = (const float*)d_in[2];
    const float* b      = (const float*)d_in[3];
    const float* alpha  = (const float*)d_in[4];
    const float* gamma  = (const float*)d_in[5];
    const float* beta   = (const float*)d_in[6];
    const float* dW     = (const float*)d_in[7];
    const float* db     = (const float*)d_in[8];
    const float* dgamma = (const float*)d_in[9];
    const float* dbeta  = (const float*)d_in[10];

    float* y      = (float*)d_out;          // pre-BN y, overwritten in-place by k_final
    float* ws     = (float*)d_ws;
    float* gstats = ws;                     // 512 floats
    float* coef   = ws + 512;               // 512 floats
    float* r_ws   = ws + 1024;              // 61,440,000 floats (pre-BN residual)

    k_zero<<<1, 256, 0, stream>>>(gstats);
    k_main<<<64*75, 256, LDS_BYTES, stream>>>(x, A, W, b, alpha, dW, db,
                                              y, r_ws, gstats);
    k_stats<<<1, 128, 0, stream>>>(gstats, gamma, beta, dgamma, dbeta, coef);
    k_final<<<60000, 256, 0, stream>>>(y, r_ws, coef);
}